// GNN_16389595201742
// MI455X (gfx1250) — compile-verified
//
#include <hip/hip_runtime.h>
#include <hip/hip_bf16.h>

typedef __attribute__((ext_vector_type(2))) float v2f;
typedef __attribute__((ext_vector_type(8))) float v8f;

#define IN_F 64
#define HID_F 64
#define OUT_F 16
#define SW_STRIDE 80   // LDS row stride (floats): +2 rows => +160 ≡ +32 banks, halves disjoint

// ---------------- degree / norm ----------------

__global__ void k_deg_init(float* __restrict__ deg, int N) {
    int i = blockIdx.x * blockDim.x + threadIdx.x;
    if (i < N) deg[i] = 1.0f;  // self loop
}

__global__ void k_deg_edges(float* __restrict__ deg, const int* __restrict__ dst, int E) {
    int e = blockIdx.x * blockDim.x + threadIdx.x;
    if (e < E) atomicAdd(&deg[dst[e]], 1.0f);
}

__global__ void k_rsqrt(float* __restrict__ deg, int N) {
    int i = blockIdx.x * blockDim.x + threadIdx.x;
    if (i < N) deg[i] = rsqrtf(deg[i]);   // deg -> dinv in place
}

__global__ void k_norm(float* __restrict__ norm, const int* __restrict__ src,
                       const int* __restrict__ dst, const float* __restrict__ dinv, int E) {
    int e = blockIdx.x * blockDim.x + threadIdx.x;
    if (e < E) norm[e] = dinv[src[e]] * dinv[dst[e]];
}

// ---------------- GEMM1: h = x @ W1  (fp32 WMMA) ----------------
// 256 threads = 8 waves; wave w handles 16 rows; block handles 128 rows.

__global__ void __launch_bounds__(256) k_gemm1(const float* __restrict__ x,
                                               const float* __restrict__ W1,
                                               float* __restrict__ h, int N) {
    __shared__ float sW[HID_F * SW_STRIDE];
    for (int i = threadIdx.x; i < IN_F * HID_F; i += 256) {
        int r = i >> 6, cc = i & 63;
        sW[r * SW_STRIDE + cc] = W1[i];
    }
    __syncthreads();

    const int wave = threadIdx.x >> 5;
    const int lane = threadIdx.x & 31;
    const int half = lane >> 4;
    const int l    = lane & 15;
    const int rowBase = (blockIdx.x * 8 + wave) * 16;
    const int row  = rowBase + l;
    const int rowc = row < N ? row : N - 1;

    // A fragments: 16x4 fp32 tiles. lane<16 holds K={0,1}, lane>=16 holds K={2,3} of each group.
    v2f a[16];
    const float* xr = x + (size_t)rowc * IN_F + 2 * half;
#pragma unroll
    for (int kk = 0; kk < 16; ++kk) {
        a[kk].x = xr[4 * kk + 0];
        a[kk].y = xr[4 * kk + 1];
    }

#pragma unroll
    for (int nt = 0; nt < 4; ++nt) {
        const int n0 = nt * 16;
        v8f acc = {};
#pragma unroll
        for (int kk = 0; kk < 16; ++kk) {
            const int k = 4 * kk + 2 * half;
            v2f b;
            b.x = sW[k * SW_STRIDE + n0 + l];
            b.y = sW[(k + 1) * SW_STRIDE + n0 + l];
            acc = __builtin_amdgcn_wmma_f32_16x16x4_f32(false, a[kk], false, b,
                                                        (short)0, acc, false, false);
        }
#pragma unroll
        for (int r = 0; r < 8; ++r) {
            const int orow = rowBase + r + 8 * half;
            if (orow < N) h[(size_t)orow * HID_F + n0 + l] = acc[r];
        }
    }
}

// ---------------- layer-1 aggregation ----------------
// agg1[i][f] = b1[f] + dinv[i]^2 * h[i][f]   (self-loop term folded in)

__global__ void k_agg1_init(float* __restrict__ agg1, const float* __restrict__ h,
                            const float* __restrict__ b1, const float* __restrict__ dinv,
                            int N) {
    int gid = blockIdx.x * blockDim.x + threadIdx.x;
    if (gid < N * HID_F) {
        int i = gid >> 6, f = gid & 63;
        float di = dinv[i];
        agg1[gid] = b1[f] + di * di * h[gid];
    }
}

// 64 threads per edge: coalesced gather of h[src][:] and coalesced atomics to agg1[dst][:]
__global__ void k_agg1_scatter(float* __restrict__ agg1, const float* __restrict__ h,
                               const float* __restrict__ norm, const int* __restrict__ src,
                               const int* __restrict__ dst, int E) {
    long long gid = (long long)blockIdx.x * blockDim.x + threadIdx.x;
    int e = (int)(gid >> 6);
    int f = (int)(gid & 63);
    if (e < E) {
        int s = src[e], d = dst[e];
        float v = h[(size_t)s * HID_F + f] * norm[e];
        atomicAdd(&agg1[(size_t)d * HID_F + f], v);
    }
}

// ---------------- GEMM2: h2 = relu(agg1) @ W2  (fp32 WMMA, relu fused into A load) ----------------

__global__ void __launch_bounds__(256) k_gemm2(const float* __restrict__ agg1,
                                               const float* __restrict__ W2,
                                               float* __restrict__ h2, int N) {
    __shared__ float sW[HID_F * SW_STRIDE];
    for (int i = threadIdx.x; i < HID_F * OUT_F; i += 256) {
        int r = i >> 4, cc = i & 15;
        sW[r * SW_STRIDE + cc] = W2[i];
    }
    __syncthreads();

    const int wave = threadIdx.x >> 5;
    const int lane = threadIdx.x & 31;
    const int half = lane >> 4;
    const int l    = lane & 15;
    const int rowBase = (blockIdx.x * 8 + wave) * 16;
    const int row  = rowBase + l;
    const int rowc = row < N ? row : N - 1;

    v2f a[16];
    const float* xr = agg1 + (size_t)rowc * HID_F + 2 * half;
#pragma unroll
    for (int kk = 0; kk < 16; ++kk) {
        a[kk].x = fmaxf(xr[4 * kk + 0], 0.0f);
        a[kk].y = fmaxf(xr[4 * kk + 1], 0.0f);
    }

    v8f acc = {};
#pragma unroll
    for (int kk = 0; kk < 16; ++kk) {
        const int k = 4 * kk + 2 * half;
        v2f b;
        b.x = sW[k * SW_STRIDE + l];
        b.y = sW[(k + 1) * SW_STRIDE + l];
        acc = __builtin_amdgcn_wmma_f32_16x16x4_f32(false, a[kk], false, b,
                                                    (short)0, acc, false, false);
    }
#pragma unroll
    for (int r = 0; r < 8; ++r) {
        const int orow = rowBase + r + 8 * half;
        if (orow < N) h2[(size_t)orow * OUT_F + l] = acc[r];
    }
}

// ---------------- layer-2 aggregation (into d_out) ----------------

__global__ void k_agg2_init(float* __restrict__ out, const float* __restrict__ h2,
                            const float* __restrict__ b2, const float* __restrict__ dinv,
                            int N) {
    int gid = blockIdx.x * blockDim.x + threadIdx.x;
    if (gid < N * OUT_F) {
        int i = gid >> 4, o = gid & 15;
        float di = dinv[i];
        out[gid] = b2[o] + di * di * h2[gid];
    }
}

// 16 threads per edge
__global__ void k_agg2_scatter(float* __restrict__ out, const float* __restrict__ h2,
                               const float* __restrict__ norm, const int* __restrict__ src,
                               const int* __restrict__ dst, int E) {
    long long gid = (long long)blockIdx.x * blockDim.x + threadIdx.x;
    int e = (int)(gid >> 4);
    int o = (int)(gid & 15);
    if (e < E) {
        int s = src[e], d = dst[e];
        float v = h2[(size_t)s * OUT_F + o] * norm[e];
        atomicAdd(&out[(size_t)d * OUT_F + o], v);
    }
}

// ---------------- log-softmax over 16 classes (in place on d_out) ----------------

__global__ void k_logsoftmax(float* __restrict__ out, int N) {
    int i = blockIdx.x * blockDim.x + threadIdx.x;
    if (i >= N) return;
    float v[OUT_F];
    float* p = out + (size_t)i * OUT_F;
#pragma unroll
    for (int o = 0; o < OUT_F; ++o) v[o] = p[o];
    float m = v[0];
#pragma unroll
    for (int o = 1; o < OUT_F; ++o) m = fmaxf(m, v[o]);
    float s = 0.0f;
#pragma unroll
    for (int o = 0; o < OUT_F; ++o) s += __expf(v[o] - m);
    float ls = m + __logf(s);
#pragma unroll
    for (int o = 0; o < OUT_F; ++o) p[o] = v[o] - ls;
}

// ---------------- host entry ----------------

extern "C" void kernel_launch(void* const* d_in, const int* in_sizes, int n_in,
                              void* d_out, int out_size, void* d_ws, size_t ws_size,
                              hipStream_t stream) {
    const float* x  = (const float*)d_in[0];
    const int*   ei = (const int*)d_in[1];
    const float* W1 = (const float*)d_in[2];
    const float* b1 = (const float*)d_in[3];
    const float* W2 = (const float*)d_in[4];
    const float* b2 = (const float*)d_in[5];

    const int N = in_sizes[0] / IN_F;
    const int E = in_sizes[1] / 2;
    const int* src = ei;
    const int* dst = ei + E;

    float* out = (float*)d_out;

    // workspace layout (floats): dinv[N] | norm[E] | h[N*64] | agg1[N*64] | h2[N*16]
    float* ws   = (float*)d_ws;
    float* dinv = ws;
    float* norm = dinv + N;
    float* h    = norm + E;
    float* agg1 = h + (size_t)N * HID_F;
    float* h2   = agg1 + (size_t)N * HID_F;

    const int TB = 256;
    const int gN    = (N + TB - 1) / TB;
    const int gE    = (E + TB - 1) / TB;
    const int gGemm = (N + 127) / 128;
    const int gA1i  = ((size_t)N * HID_F + TB - 1) / TB;
    const int gA1s  = (int)(((long long)E * HID_F + TB - 1) / TB);
    const int gA2i  = ((size_t)N * OUT_F + TB - 1) / TB;
    const int gA2s  = (int)(((long long)E * OUT_F + TB - 1) / TB);

    k_deg_init    <<<gN,    TB, 0, stream>>>(dinv, N);
    k_deg_edges   <<<gE,    TB, 0, stream>>>(dinv, dst, E);
    k_rsqrt       <<<gN,    TB, 0, stream>>>(dinv, N);
    k_norm        <<<gE,    TB, 0, stream>>>(norm, src, dst, dinv, E);

    k_gemm1       <<<gGemm, TB, 0, stream>>>(x, W1, h, N);
    k_agg1_init   <<<gA1i,  TB, 0, stream>>>(agg1, h, b1, dinv, N);
    k_agg1_scatter<<<gA1s,  TB, 0, stream>>>(agg1, h, norm, src, dst, E);

    k_gemm2       <<<gGemm, TB, 0, stream>>>(agg1, W2, h2, N);
    k_agg2_init   <<<gA2i,  TB, 0, stream>>>(out, h2, b2, dinv, N);
    k_agg2_scatter<<<gA2s,  TB, 0, stream>>>(out, h2, norm, src, dst, E);

    k_logsoftmax  <<<gN,    TB, 0, stream>>>(out, N);
}